// MultiHeadAttention_44744969289805
// MI455X (gfx1250) — compile-verified
//
#include <hip/hip_runtime.h>
#include <hip/hip_bf16.h>

typedef float v2f __attribute__((ext_vector_type(2)));
typedef float v8f __attribute__((ext_vector_type(8)));

#define D_MODEL 1024
#define HEADS   16
#define DK      64
#define BATCH   2
#define SEQ     2048

// D = A(16x4,f32) * B(4x16,f32) + C(16x16,f32) on the CDNA5 matrix pipe.
__device__ __forceinline__ v8f wmma4(v2f a, v2f b, v8f c) {
  return __builtin_amdgcn_wmma_f32_16x16x4_f32(false, a, false, b, (short)0, c,
                                               false, false);
}

__device__ __forceinline__ v2f ld2g(const float* __restrict__ p) {
  return *(const v2f* __restrict__)p;
}

// C[M,N] = A[M,K] @ W[N,K]^T + bias[N]   (torch Linear)
// One wave per block computes a 32x64 output strip (2x4 WMMA tiles).
// Two fragment stages (k and k+4) are kept in flight; each stage is refilled
// a full iteration (8 WMMAs) before it is consumed, so loadcnt waits only
// cover older loads and the matrix pipe stays busy.
__global__ __launch_bounds__(32) void gemm_wt_bias(
    const float* __restrict__ A, const float* __restrict__ W,
    const float* __restrict__ bias, float* __restrict__ C,
    int M, int N, int K) {
  const int lane = threadIdx.x;
  const int lm   = lane & 15;
  const int half = lane >> 4;
  const int m0   = blockIdx.y << 5;   // 32 rows
  const int n0   = blockIdx.x << 6;   // 64 cols

  v8f acc[2][4];
#pragma unroll
  for (int i = 0; i < 2; ++i)
#pragma unroll
    for (int t = 0; t < 4; ++t)
#pragma unroll
      for (int r = 0; r < 8; ++r) acc[i][t][r] = 0.f;

  const float* ap0 = A + (size_t)(m0 + lm) * K + (half << 1);
  const float* ap1 = ap0 + (size_t)16 * K;
  const float* wp  = W + (size_t)(n0 + lm) * K + (half << 1);
  const size_t wstep = (size_t)16 * K;

#define LOAD_STAGE(A0, A1, B0, B1, B2, B3, OFF)                                \
  A0 = ld2g(ap0 + (OFF));                                                      \
  A1 = ld2g(ap1 + (OFF));                                                      \
  B0 = ld2g(wp + (OFF));                                                       \
  B1 = ld2g(wp + wstep + (OFF));                                               \
  B2 = ld2g(wp + 2 * wstep + (OFF));                                           \
  B3 = ld2g(wp + 3 * wstep + (OFF));

#define GEMM_STEP(A0, A1, B0, B1, B2, B3)                                      \
  acc[0][0] = wmma4(A0, B0, acc[0][0]);                                        \
  acc[1][0] = wmma4(A1, B0, acc[1][0]);                                        \
  acc[0][1] = wmma4(A0, B1, acc[0][1]);                                        \
  acc[1][1] = wmma4(A1, B1, acc[1][1]);                                        \
  acc[0][2] = wmma4(A0, B2, acc[0][2]);                                        \
  acc[1][2] = wmma4(A1, B2, acc[1][2]);                                        \
  acc[0][3] = wmma4(A0, B3, acc[0][3]);                                        \
  acc[1][3] = wmma4(A1, B3, acc[1][3]);

  // stage A <- k=0, stage B <- k=4
  v2f xa0, xa1, xb0, xb1, xb2, xb3;
  v2f ya0, ya1, yb0, yb1, yb2, yb3;
  LOAD_STAGE(xa0, xa1, xb0, xb1, xb2, xb3, 0)
  LOAD_STAGE(ya0, ya1, yb0, yb1, yb2, yb3, 4)

  for (int kk = 0; kk + 16 <= K; kk += 8) {
    GEMM_STEP(xa0, xa1, xb0, xb1, xb2, xb3)           // compute k = kk
    LOAD_STAGE(xa0, xa1, xb0, xb1, xb2, xb3, kk + 8)  // refill for kk+8
    GEMM_STEP(ya0, ya1, yb0, yb1, yb2, yb3)           // compute k = kk+4
    LOAD_STAGE(ya0, ya1, yb0, yb1, yb2, yb3, kk + 12) // refill for kk+12
  }
  // epilogue: stages hold K-8 and K-4
  GEMM_STEP(xa0, xa1, xb0, xb1, xb2, xb3)
  GEMM_STEP(ya0, ya1, yb0, yb1, yb2, yb3)

#undef LOAD_STAGE
#undef GEMM_STEP

#pragma unroll
  for (int i = 0; i < 2; ++i)
#pragma unroll
    for (int t = 0; t < 4; ++t) {
      const int n  = n0 + (t << 4) + lm;
      const float bv = bias[n];
#pragma unroll
      for (int r = 0; r < 8; ++r) {
        const int m = m0 + (i << 4) + r + (half << 3);  // C/D: M = r + 8*half
        C[(size_t)m * N + n] = acc[i][t][r] + bv;
      }
    }
}

// Flash attention, causal. One wave per block owns one 16-row query tile of
// one (batch, head). Q/K/V/O are [B, S, D_MODEL] with head offset h*DK.
// Next key tile's K-fragments are prefetched right after the current QK^T
// WMMAs so the loads overlap softmax + PV; V-fragments are batch-loaded.
__global__ __launch_bounds__(32) void flash_attn(
    const float* __restrict__ Q, const float* __restrict__ Km,
    const float* __restrict__ V, float* __restrict__ O, int S) {
  __shared__ __align__(16) float sP[16 * 18];  // prob tile, padded stride
  __shared__ float row_m[16];
  __shared__ float row_l[16];
  __shared__ float row_sc[16];

  const int lane = threadIdx.x;
  const int lm   = lane & 15;
  const int half = lane >> 4;
  const int m0   = blockIdx.x << 4;
  const int b    = blockIdx.y >> 4;   // HEADS == 16
  const int h    = blockIdx.y & 15;

  const size_t headOff = (size_t)b * S * D_MODEL + (size_t)h * DK;
  const float* Qb = Q + headOff;
  const float* Kb = Km + headOff;
  const float* Vb = V + headOff;

  // Preload the 16x64 Q tile as 16 A-fragments.
  v2f qf[16];
  {
    const float* qp = Qb + (size_t)(m0 + lm) * D_MODEL + (half << 1);
#pragma unroll
    for (int kb = 0; kb < 16; ++kb) qf[kb] = ld2g(qp + (kb << 2));
  }

  v8f o[4];
#pragma unroll
  for (int t = 0; t < 4; ++t)
#pragma unroll
    for (int r = 0; r < 8; ++r) o[t][r] = 0.f;

  if (lane < 16) {
    row_m[lane] = -3.0e38f;
    row_l[lane] = 0.f;
  }
  __syncthreads();

  // K-fragments for tile 0.
  v2f kf[16];
  {
    const float* kp = Kb + (size_t)lm * D_MODEL + (half << 1);
#pragma unroll
    for (int kb = 0; kb < 16; ++kb) kf[kb] = ld2g(kp + (kb << 2));
  }

  const int ntiles = (m0 >> 4) + 1;  // causal: key tiles up to the diagonal
  for (int nt = 0; nt < ntiles; ++nt) {
    const int n0 = nt << 4;

    // ---- S = Q . K^T  (16 WMMAs, K=64) ----
    v8f s;
#pragma unroll
    for (int r = 0; r < 8; ++r) s[r] = 0.f;
#pragma unroll
    for (int kb = 0; kb < 16; ++kb) s = wmma4(qf[kb], kf[kb], s);

    // ---- prefetch next tile's K-fragments (overlaps softmax + PV) ----
    if (nt + 1 < ntiles) {
      const float* kp = Kb + (size_t)(n0 + 16 + lm) * D_MODEL + (half << 1);
#pragma unroll
      for (int kb = 0; kb < 16; ++kb) kf[kb] = ld2g(kp + (kb << 2));
    }

    // ---- scale, causal mask, stage tile in LDS ----
    const int n_g = n0 + lm;
#pragma unroll
    for (int r = 0; r < 8; ++r) {
      const int m_g = m0 + r + (half << 3);
      float val = s[r] * 0.125f;            // 1/sqrt(64)
      val = (n_g > m_g) ? -1.0e30f : val;   // causal tril mask
      sP[(r + (half << 3)) * 18 + lm] = val;
    }
    __syncthreads();

    // ---- online softmax: lanes 0..15 each own one row ----
    if (lane < 16) {
      const float mprev = row_m[lane];
      float tmax = mprev;
#pragma unroll
      for (int c = 0; c < 16; ++c) tmax = fmaxf(tmax, sP[lane * 18 + c]);
      const float mnew = tmax;
      const float sc   = __expf(mprev - mnew);
      float tsum = 0.f;
#pragma unroll
      for (int c = 0; c < 16; ++c) {
        const float p = __expf(sP[lane * 18 + c] - mnew);
        sP[lane * 18 + c] = p;
        tsum += p;
      }
      row_l[lane]  = row_l[lane] * sc + tsum;
      row_m[lane]  = mnew;
      row_sc[lane] = sc;
    }
    __syncthreads();

    // ---- batch-load V fragments: B(k,n) = V[n0+k][16t+n] ----
    v2f vb[16];
#pragma unroll
    for (int kb = 0; kb < 4; ++kb) {
      const int k0 = n0 + (kb << 2) + (half << 1);
#pragma unroll
      for (int t = 0; t < 4; ++t) {
        const float* vp = Vb + (size_t)k0 * D_MODEL + (t << 4) + lm;
        vb[(kb << 2) + t][0] = vp[0];
        vb[(kb << 2) + t][1] = vp[D_MODEL];
      }
    }

    // ---- rescale running O by exp(m_old - m_new) ----
#pragma unroll
    for (int r = 0; r < 8; ++r) {
      const float f = row_sc[r + (half << 3)];
#pragma unroll
      for (int t = 0; t < 4; ++t) o[t][r] *= f;
    }

    // ---- O += P . V  (16 WMMAs, K=16) ----
#pragma unroll
    for (int kb = 0; kb < 4; ++kb) {
      // A-frag of P from LDS: row = lane%16, k = 4*kb + 2*half + j (aligned)
      v2f a = *(const v2f*)&sP[lm * 18 + (kb << 2) + (half << 1)];
#pragma unroll
      for (int t = 0; t < 4; ++t) o[t] = wmma4(a, vb[(kb << 2) + t], o[t]);
    }
    __syncthreads();  // sP / row_sc reused next iteration
  }

  // ---- normalize by l and store ----
#pragma unroll
  for (int r = 0; r < 8; ++r) {
    const float invl = 1.f / row_l[r + (half << 3)];
#pragma unroll
    for (int t = 0; t < 4; ++t) o[t][r] *= invl;
  }
#pragma unroll
  for (int t = 0; t < 4; ++t)
#pragma unroll
    for (int r = 0; r < 8; ++r) {
      const int m = m0 + r + (half << 3);
      O[headOff + (size_t)m * D_MODEL + (t << 4) + lm] = o[t][r];
    }
}

extern "C" void kernel_launch(void* const* d_in, const int* in_sizes, int n_in,
                              void* d_out, int out_size, void* d_ws,
                              size_t ws_size, hipStream_t stream) {
  const float* q   = (const float*)d_in[0];
  const float* k   = (const float*)d_in[1];
  const float* v   = (const float*)d_in[2];
  // d_in[3] is the causal tril mask; applied analytically in flash_attn.
  const float* w_q = (const float*)d_in[4];
  const float* b_q = (const float*)d_in[5];
  const float* w_k = (const float*)d_in[6];
  const float* b_k = (const float*)d_in[7];
  const float* w_v = (const float*)d_in[8];
  const float* b_v = (const float*)d_in[9];
  const float* w_o = (const float*)d_in[10];
  const float* b_o = (const float*)d_in[11];

  const int M = BATCH * SEQ;  // 4096 rows
  float* Qp = (float*)d_ws;
  float* Kp = Qp + (size_t)M * D_MODEL;
  float* Vp = Kp + (size_t)M * D_MODEL;
  float* Ob = Vp + (size_t)M * D_MODEL;   // 4 x 16 MB workspace

  dim3 gp(D_MODEL / 64, M / 32);          // (16, 128) waves
  gemm_wt_bias<<<gp, 32, 0, stream>>>(q, w_q, b_q, Qp, M, D_MODEL, D_MODEL);
  gemm_wt_bias<<<gp, 32, 0, stream>>>(k, w_k, b_k, Kp, M, D_MODEL, D_MODEL);
  gemm_wt_bias<<<gp, 32, 0, stream>>>(v, w_v, b_v, Vp, M, D_MODEL, D_MODEL);

  dim3 ga(SEQ / 16, BATCH * HEADS);       // (128, 32) waves
  flash_attn<<<ga, 32, 0, stream>>>(Qp, Kp, Vp, Ob, SEQ);

  gemm_wt_bias<<<gp, 32, 0, stream>>>(Ob, w_o, b_o, (float*)d_out,
                                      M, D_MODEL, D_MODEL);
}